// snn_mnist_86449101734047
// MI455X (gfx1250) — compile-verified
//
#include <hip/hip_runtime.h>

// SNN + STDP over time, decomposed per neuron row.
// 32 blocks x 16 neurons; 7 waves/block, each wave owns a 128-col tile of the
// 16x784 weight block resident in LDS. Dot products via V_WMMA_F32_16X16X4_F32
// with x broadcast across all 16 B-columns; STDP rank-1 update + clamp applied
// in LDS and streamed to W_rec with b128 stores.

typedef __attribute__((ext_vector_type(2))) float v2f;
typedef __attribute__((ext_vector_type(8))) float v8f;

#define NSTEPS 256
#define DIN    784
#define NNEUR  512
#define NB     16                 // neurons per block (one WMMA M-block)
#define NWAVES 7                  // 7 x 128 = 896 padded columns
#define TPB    (NWAVES * 32)
#define WSTR   900                // LDS row stride (words): conflict-free, 16B aligned

#define ALPHA      0.9f
#define BETA       0.8f
#define THRESH     1.0f
#define BETA_MINUS 0.9f
#define BETA_PLUS  0.9f
#define A_PLUS     0.008f
#define A_MINUS    0.0066f

__global__ __launch_bounds__(TPB, 1)
void snn_stdp_kernel(const float* __restrict__ img,   // [256,784]
                     const float* __restrict__ W0,    // [512,784]
                     float* __restrict__ out)
{
    __shared__ __align__(16) float Wl[NB * WSTR];     // 57,600 B
    __shared__ __align__(16) float xl[NWAVES * 128];  //  3,584 B (zero-padded x_t)
    __shared__ __align__(16) float pl[NWAVES * 128];  //  3,584 B (pre trace)
    __shared__ __align__(16) float partial[NWAVES * 16];
    __shared__ __align__(16) float apv[16];
    __shared__ __align__(16) float amv[16];

    const int tid   = threadIdx.x;
    const int lane  = tid & 31;
    const int wv    = tid >> 5;
    const int m     = lane & 15;          // WMMA row (A) / column (D)
    const int half  = lane >> 4;          // 0: K=0,1  1: K=2,3
    const int aoff  = half * 2;
    const int nbase = blockIdx.x * NB;
    const int tbase = wv * 128;           // this wave's column tile
    const int k0    = tbase + 4 * lane;   // 4 consecutive cols per lane
    const bool kval = (k0 + 3) < DIN;

    float* outW    = out;
    float* outSpk  = out + (size_t)NSTEPS * NNEUR * DIN;
    float* outMem  = outSpk + (size_t)NSTEPS * NNEUR;
    float* outSyn  = outMem + (size_t)NSTEPS * NNEUR;
    float* outPost = outSyn + (size_t)NSTEPS * NNEUR;

    // ---- init: stage W block into LDS (pads -> 0), zero pre trace ----
    for (int n = 0; n < NB; ++n) {
        float4 w4 = make_float4(0.f, 0.f, 0.f, 0.f);
        if (kval) w4 = *(const float4*)(W0 + (size_t)(nbase + n) * DIN + k0);
        *(float4*)&Wl[n * WSTR + k0] = w4;
    }
    *(float4*)&pl[k0] = make_float4(0.f, 0.f, 0.f, 0.f);

    float syn = 0.f, mem = 0.f, spk = 0.f, post = 0.f;  // live in wave0 lanes 0-15
    __syncthreads();

    for (int t = 0; t < NSTEPS; ++t) {
        // ---- phase 0: load x_t tile (pads stay 0 so WMMA tail contributes 0)
        float4 xv = make_float4(0.f, 0.f, 0.f, 0.f);
        if (kval) xv = *(const float4*)(img + (size_t)t * DIN + k0);
        *(float4*)&xl[k0] = xv;

        // ---- phase 1: partial dots for 16 neurons over this wave's 128 cols
        v8f acc = {};
        #pragma unroll
        for (int c = 0; c < 32; ++c) {
            const int col = tbase + 4 * c + aoff;
            v2f a = *(const v2f*)&Wl[m * WSTR + col];  // A: W[m][k], k=col..col+1
            v2f b = *(const v2f*)&xl[col];             // B: x[k] broadcast over N
            acc = __builtin_amdgcn_wmma_f32_16x16x4_f32(
                false, a, false, b, (short)0, acc, false, false);
        }
        // D col 0 == all cols; lane 0 holds rows 0-7, lane 16 rows 8-15
        if (m == 0) {
            float* p = &partial[wv * 16 + half * 8];
            #pragma unroll
            for (int r = 0; r < 8; ++r) p[r] = acc[r];
        }
        __syncthreads();

        // ---- phase 2: LIF + traces (wave0, one lane per neuron)
        if (wv == 0 && lane < 16) {
            float I = 0.f;
            #pragma unroll
            for (int q = 0; q < NWAVES; ++q) I += partial[q * 16 + lane];
            syn = ALPHA * syn + I;
            mem = BETA * mem + syn - spk * THRESH;     // spk == previous spike
            spk = (mem > THRESH) ? 1.0f : 0.0f;
            post = BETA_MINUS * post + spk;
            apv[lane] = A_PLUS * spk;
            amv[lane] = A_MINUS * post;
            const size_t si = (size_t)t * NNEUR + nbase + lane;
            outSpk[si] = spk; outMem[si] = mem; outSyn[si] = syn; outPost[si] = post;
        }
        __syncthreads();

        // ---- phase 3: STDP update + clamp in LDS, stream W_rec[t] slice
        float4 pv = *(float4*)&pl[k0];
        #pragma unroll 4
        for (int n = 0; n < NB; ++n) {
            const float ap = apv[n];
            const float am = amv[n];
            float4 w4 = *(float4*)&Wl[n * WSTR + k0];
            w4.x = fminf(fmaxf(w4.x + ap * pv.x - am * xv.x, 0.f), 1.f);
            w4.y = fminf(fmaxf(w4.y + ap * pv.y - am * xv.y, 0.f), 1.f);
            w4.z = fminf(fmaxf(w4.z + ap * pv.z - am * xv.z, 0.f), 1.f);
            w4.w = fminf(fmaxf(w4.w + ap * pv.w - am * xv.w, 0.f), 1.f);
            *(float4*)&Wl[n * WSTR + k0] = w4;
            if (kval)
                *(float4*)(outW + ((size_t)t * NNEUR + nbase + n) * DIN + k0) = w4;
        }
        // pre-trace recurrence (after use in LTP)
        pv.x = BETA_PLUS * pv.x + xv.x;
        pv.y = BETA_PLUS * pv.y + xv.y;
        pv.z = BETA_PLUS * pv.z + xv.z;
        pv.w = BETA_PLUS * pv.w + xv.w;
        *(float4*)&pl[k0] = pv;
        // No barrier: next phase-1/0 only touches LDS this wave wrote; partial/
        // apv rewrites are fenced by this step's two barriers.
    }
}

extern "C" void kernel_launch(void* const* d_in, const int* in_sizes, int n_in,
                              void* d_out, int out_size, void* d_ws, size_t ws_size,
                              hipStream_t stream) {
    (void)in_sizes; (void)n_in; (void)d_ws; (void)ws_size; (void)out_size;
    const float* img = (const float*)d_in[0];   // [256,784] f32
    const float* W0  = (const float*)d_in[1];   // [512,784] f32
    float* out = (float*)d_out;                 // W_rec | spk | mem | syn | post
    snn_stdp_kernel<<<dim3(NNEUR / NB), dim3(TPB), 0, stream>>>(img, W0, out);
}